// SS1D_83889301225821
// MI455X (gfx1250) — compile-verified
//
#include <hip/hip_runtime.h>
#include <hip/hip_bf16.h>

typedef __attribute__((ext_vector_type(2))) float v2f;
typedef __attribute__((ext_vector_type(8))) float v8f;

#define B_ 8
#define L_ 8192
#define D_ 64
#define NSTATE 11
#define DN (D_ * NSTATE)         // 704
#define CL 64                    // conv l-tile per block
#define NBLK1 (B_ * (L_ / CL))   // 1024
#define CH 32                    // ssm chunk length
#define SEG 32                   // segments per batch
#define SEGL (L_ / SEG)          // 256 positions per segment
#define NCH (SEGL / CH)          // 8 chunks per segment
#define NBLK3 (B_ * SEG)         // 256 segment blocks

// ---------------------------------------------------------------------------
// Kernel 1: depthwise conv (k=63, pad 31) + per-block BN partial sums
// ---------------------------------------------------------------------------
__global__ __launch_bounds__(256) void ss1d_conv_kernel(
    const float* __restrict__ x, const float* __restrict__ w,
    float* __restrict__ xc, float* __restrict__ ps, float* __restrict__ pq) {
  __shared__ float xs[(CL + 62) * 64];   // halo tile, row0 <-> l0-31
  __shared__ float wl[64 * 63];
  __shared__ float rs_[256], rq_[256];

  const int tid = threadIdx.x;
  const int blk = blockIdx.x;
  const int b = blk / (L_ / CL);
  const int l0 = (blk % (L_ / CL)) * CL;

  for (int i = tid; i < (CL + 62) * 64; i += 256) {
    int r = i >> 6;
    int d = i & 63;
    int gl = l0 + r - 31;
    xs[i] = (gl >= 0 && gl < L_) ? x[((size_t)b * L_ + gl) * 64 + d] : 0.f;
  }
  for (int i = tid; i < 64 * 63; i += 256) wl[i] = w[i];
  __syncthreads();

  const int d = tid & 63;
  const int lr = tid >> 6;   // 0..3
  float s = 0.f, q = 0.f;
  for (int i = 0; i < CL / 4; ++i) {
    int li = i * 4 + lr;
    float acc = 0.f;
    for (int k = 0; k < 63; ++k)
      acc = fmaf(xs[(li + k) * 64 + d], wl[d * 63 + k], acc);
    xc[((size_t)b * L_ + l0 + li) * 64 + d] = acc;
    s += acc;
    q = fmaf(acc, acc, q);
  }
  rs_[tid] = s;
  rq_[tid] = q;
  __syncthreads();
  if (tid < 64) {
    float ss = rs_[tid] + rs_[tid + 64] + rs_[tid + 128] + rs_[tid + 192];
    float qq = rq_[tid] + rq_[tid + 64] + rq_[tid + 128] + rq_[tid + 192];
    ps[blk * 64 + tid] = ss;
    pq[blk * 64 + tid] = qq;
  }
}

// ---------------------------------------------------------------------------
// Kernel 2: BN stats finalize -> fused scale/shift per channel
// ---------------------------------------------------------------------------
__global__ void ss1d_bnfin_kernel(const float* __restrict__ ps,
                                  const float* __restrict__ pq,
                                  const float* __restrict__ gamma,
                                  const float* __restrict__ beta,
                                  float* __restrict__ stats) {
  int d = threadIdx.x;  // 64 threads
  float s = 0.f, q = 0.f;
  for (int i = 0; i < NBLK1; ++i) {
    s += ps[i * 64 + d];
    q += pq[i * 64 + d];
  }
  const float inv_n = 1.f / (float)(B_ * L_);
  float mean = s * inv_n;
  float var = q * inv_n - mean * mean;
  float inv = rsqrtf(var + 1e-5f);
  float scl = gamma[d] * inv;
  stats[d] = scl;
  stats[64 + d] = beta[d] - mean * scl;
}

// ---------------------------------------------------------------------------
// Kernel 3a (Pass A): per-segment front-end + local scan from h=0.
// Emits per (b,seg,d,n): P = exp(a*sum(delta)) and Q = local end-state.
// ---------------------------------------------------------------------------
__global__ __launch_bounds__(256) void ss1d_seg_kernel(
    const float* __restrict__ xc, const float* __restrict__ stats,
    const float* __restrict__ xpw, const float* __restrict__ dtw,
    const float* __restrict__ dtb, const float* __restrict__ alogs,
    float* __restrict__ Pw, float* __restrict__ Qw) {
  __shared__ float u_l[64][CH + 1];
  __shared__ float de_l[64][CH + 1];
  __shared__ float xd_l[32][CH + 1];
  __shared__ float wp[32][65];
  __shared__ float sdt[64][4];
  __shared__ float sbias[64], sscl[64], ssft[64];

  const int tid = threadIdx.x;
  const int b = blockIdx.x >> 5;
  const int seg = blockIdx.x & 31;
  const int lbase = seg * SEGL;

  for (int i = tid; i < 32 * 64; i += 256) {
    int c = i >> 6, d = i & 63;
    wp[c][d] = (c < 26) ? xpw[c * 64 + d] : 0.f;
  }
  sdt[tid >> 2][tid & 3] = dtw[tid];
  if (tid < 64) {
    sbias[tid] = dtb[tid];
    sscl[tid] = stats[tid];
    ssft[tid] = stats[64 + tid];
  }

  const int d_s = tid & 63;
  const int n0 = (tid >> 6) * 3;
  const bool v2 = (n0 + 2) < NSTATE;
  const float a0 = -__expf(alogs[d_s * NSTATE + n0]);
  const float a1 = -__expf(alogs[d_s * NSTATE + n0 + 1]);
  const float a2 = v2 ? -__expf(alogs[d_s * NSTATE + n0 + 2]) : 0.f;
  float h0 = 0.f, h1 = 0.f, h2 = 0.f, dsum = 0.f;

  const int wave = tid >> 5, lane = tid & 31;
  const int mr = lane & 15;
  const int khb = (lane >> 4) * 2;

  __syncthreads();

  for (int c = 0; c < NCH; ++c) {
    const int l0 = lbase + c * CH;
    if (c + 1 < NCH) {  // prefetch next chunk of xc (8 KB)
      const char* pf = (const char*)(xc + ((size_t)b * L_ + l0 + CH) * 64);
      __builtin_prefetch(pf + tid * 32, 0, 0);
    }
    // P1: BN + SiLU, transpose into LDS
    for (int i = tid; i < 64 * CH; i += 256) {
      int li = i >> 6, d = i & 63;
      float v = xc[((size_t)b * L_ + l0 + li) * 64 + d];
      v = fmaf(v, sscl[d], ssft[d]);
      u_l[d][li] = v / (1.f + __expf(-v));
    }
    __syncthreads();
    // P2: x_dbl = Wp * u via WMMA f32 16x16x4
    if (wave < 4) {
      const int mt = wave & 1, nt = wave >> 1;
      v8f acc = {0.f, 0.f, 0.f, 0.f, 0.f, 0.f, 0.f, 0.f};
#pragma unroll
      for (int k = 0; k < 64; k += 4) {
        v2f av, bv;
        av.x = wp[mt * 16 + mr][k + khb];
        av.y = wp[mt * 16 + mr][k + khb + 1];
        bv.x = u_l[k + khb][nt * 16 + mr];
        bv.y = u_l[k + khb + 1][nt * 16 + mr];
        acc = __builtin_amdgcn_wmma_f32_16x16x4_f32(
            false, av, false, bv, (short)0, acc, false, false);
      }
      const int rbase = mt * 16 + (lane >> 4) * 8;
      const int col = nt * 16 + mr;
#pragma unroll
      for (int j = 0; j < 8; ++j) xd_l[rbase + j][col] = acc[j];
    }
    __syncthreads();
    // P3: delta + du
    for (int i = tid; i < 64 * CH; i += 256) {
      int li = i >> 6, d = i & 63;
      float s = sbias[d];
#pragma unroll
      for (int r = 0; r < 4; ++r) s = fmaf(xd_l[r][li], sdt[d][r], s);
      float delta = (s > 20.f) ? s : log1pf(__expf(s));
      de_l[d][li] = delta;
      u_l[d][li] = delta * u_l[d][li];
    }
    __syncthreads();
    // P4: local scan (no y)
#pragma unroll
    for (int li = 0; li < CH; ++li) {
      float dl = de_l[d_s][li];
      float du = u_l[d_s][li];
      dsum += dl;
      h0 = fmaf(__expf(dl * a0), h0, du * xd_l[4 + n0][li]);
      h1 = fmaf(__expf(dl * a1), h1, du * xd_l[5 + n0][li]);
      if (v2) h2 = fmaf(__expf(dl * a2), h2, du * xd_l[6 + n0][li]);
    }
    __syncthreads();
  }

  const size_t o = ((size_t)blockIdx.x) * DN + d_s * NSTATE + n0;
  Pw[o] = __expf(a0 * dsum);
  Qw[o] = h0;
  Pw[o + 1] = __expf(a1 * dsum);
  Qw[o + 1] = h1;
  if (v2) {
    Pw[o + 2] = __expf(a2 * dsum);
    Qw[o + 2] = h2;
  }
}

// ---------------------------------------------------------------------------
// Kernel 3b (Pass B): combine segment summaries -> initial state per segment
// ---------------------------------------------------------------------------
__global__ __launch_bounds__(256) void ss1d_comb_kernel(
    const float* __restrict__ Pw, const float* __restrict__ Qw,
    float* __restrict__ hinit) {
  const int idx = blockIdx.x * 256 + threadIdx.x;  // 22*256 = 5632 = B_*DN
  const int b = idx / DN;
  const int r = idx % DN;
  float H = 0.f;
  for (int s = 0; s < SEG; ++s) {
    size_t o = ((size_t)(b * SEG + s)) * DN + r;
    hinit[o] = H;
    H = fmaf(Pw[o], H, Qw[o]);
  }
}

// ---------------------------------------------------------------------------
// Kernel 3c (Pass C): front-end again + seeded scan + y + RMS + gate + out
// ---------------------------------------------------------------------------
__global__ __launch_bounds__(256) void ss1d_ssm_kernel(
    const float* __restrict__ x, const float* __restrict__ xc,
    const float* __restrict__ stats, const float* __restrict__ xpw,
    const float* __restrict__ dtw, const float* __restrict__ dtb,
    const float* __restrict__ alogs, const float* __restrict__ Dsp,
    const float* __restrict__ rsp, const float* __restrict__ hinit,
    float* __restrict__ out) {
  __shared__ float u_l[64][CH + 1];
  __shared__ float de_l[64][CH + 1];
  __shared__ float xd_l[32][CH + 1];
  __shared__ float y_l[64][CH + 1];
  __shared__ float wp[32][65];
  __shared__ float sdt[64][4];
  __shared__ float sbias[64], sDs[64], srs[64], sscl[64], ssft[64];
  __shared__ float sinv[CH];

  const int tid = threadIdx.x;
  const int b = blockIdx.x >> 5;
  const int lbase = (blockIdx.x & 31) * SEGL;

  for (int i = tid; i < 32 * 64; i += 256) {
    int c = i >> 6, d = i & 63;
    wp[c][d] = (c < 26) ? xpw[c * 64 + d] : 0.f;
  }
  sdt[tid >> 2][tid & 3] = dtw[tid];
  if (tid < 64) {
    sbias[tid] = dtb[tid];
    sDs[tid] = Dsp[tid];
    srs[tid] = rsp[tid];
    sscl[tid] = stats[tid];
    ssft[tid] = stats[64 + tid];
  }

  const int d_s = tid & 63;
  const int grp = tid >> 6;
  const int n0 = grp * 3;
  const bool v2 = (n0 + 2) < NSTATE;
  const float a0 = -__expf(alogs[d_s * NSTATE + n0]);
  const float a1 = -__expf(alogs[d_s * NSTATE + n0 + 1]);
  const float a2 = v2 ? -__expf(alogs[d_s * NSTATE + n0 + 2]) : 0.f;
  const size_t ho = ((size_t)blockIdx.x) * DN + d_s * NSTATE + n0;
  float h0 = hinit[ho];
  float h1 = hinit[ho + 1];
  float h2 = v2 ? hinit[ho + 2] : 0.f;

  const int wave = tid >> 5, lane = tid & 31;
  const int mr = lane & 15;
  const int khb = (lane >> 4) * 2;

  __syncthreads();

  for (int c = 0; c < NCH; ++c) {
    const int l0 = lbase + c * CH;
    if (c + 1 < NCH) {  // prefetch next chunk of xc and x (8 KB each)
      const char* pf = (const char*)(xc + ((size_t)b * L_ + l0 + CH) * 64);
      __builtin_prefetch(pf + tid * 32, 0, 0);
      const char* pz = (const char*)(x + ((size_t)b * L_ + l0 + CH) * 64);
      __builtin_prefetch(pz + tid * 32, 0, 0);
    }
    // P1: BN + SiLU, transpose into LDS
    for (int i = tid; i < 64 * CH; i += 256) {
      int li = i >> 6, d = i & 63;
      float v = xc[((size_t)b * L_ + l0 + li) * 64 + d];
      v = fmaf(v, sscl[d], ssft[d]);
      u_l[d][li] = v / (1.f + __expf(-v));
    }
    __syncthreads();
    // P2: x_dbl = Wp * u via WMMA f32 16x16x4
    if (wave < 4) {
      const int mt = wave & 1, nt = wave >> 1;
      v8f acc = {0.f, 0.f, 0.f, 0.f, 0.f, 0.f, 0.f, 0.f};
#pragma unroll
      for (int k = 0; k < 64; k += 4) {
        v2f av, bv;
        av.x = wp[mt * 16 + mr][k + khb];
        av.y = wp[mt * 16 + mr][k + khb + 1];
        bv.x = u_l[k + khb][nt * 16 + mr];
        bv.y = u_l[k + khb + 1][nt * 16 + mr];
        acc = __builtin_amdgcn_wmma_f32_16x16x4_f32(
            false, av, false, bv, (short)0, acc, false, false);
      }
      const int rbase = mt * 16 + (lane >> 4) * 8;
      const int col = nt * 16 + mr;
#pragma unroll
      for (int j = 0; j < 8; ++j) xd_l[rbase + j][col] = acc[j];
    }
    __syncthreads();
    // P3: delta, y init, du
    for (int i = tid; i < 64 * CH; i += 256) {
      int li = i >> 6, d = i & 63;
      float s = sbias[d];
#pragma unroll
      for (int r = 0; r < 4; ++r) s = fmaf(xd_l[r][li], sdt[d][r], s);
      float delta = (s > 20.f) ? s : log1pf(__expf(s));
      float uu = u_l[d][li];
      de_l[d][li] = delta;
      y_l[d][li] = sDs[d] * uu;
      u_l[d][li] = delta * uu;
    }
    __syncthreads();
    // P4: seeded scan
    float yacc[CH];
#pragma unroll
    for (int li = 0; li < CH; ++li) {
      float dl = de_l[d_s][li];
      float du = u_l[d_s][li];
      h0 = fmaf(__expf(dl * a0), h0, du * xd_l[4 + n0][li]);
      float yp = h0 * xd_l[15 + n0][li];
      h1 = fmaf(__expf(dl * a1), h1, du * xd_l[5 + n0][li]);
      yp = fmaf(h1, xd_l[16 + n0][li], yp);
      if (v2) {
        h2 = fmaf(__expf(dl * a2), h2, du * xd_l[6 + n0][li]);
        yp = fmaf(h2, xd_l[17 + n0][li], yp);
      }
      yacc[li] = yp;
    }
    // deterministic 4-pass reduction of the state groups into y_l
    for (int g = 0; g < 4; ++g) {
      if (grp == g) {
#pragma unroll
        for (int li = 0; li < CH; ++li) y_l[d_s][li] += yacc[li];
      }
      __syncthreads();
    }
    // P5: RMS over channels per position
    if (tid < CH) {
      float s = 0.f;
      for (int d2 = 0; d2 < 64; ++d2) {
        float v = y_l[d2][tid];
        s = fmaf(v, v, s);
      }
      sinv[tid] = rsqrtf(s * (1.f / 64.f) + 1e-5f);
    }
    __syncthreads();
    // P6: RMS-scale + SiLU(z) gate, coalesced store
    for (int i = tid; i < 64 * CH; i += 256) {
      int li = i >> 6, d = i & 63;
      size_t gi = ((size_t)b * L_ + l0 + li) * 64 + d;
      float zv = x[gi];
      float sz = zv / (1.f + __expf(-zv));
      out[gi] = y_l[d][li] * srs[d] * sinv[li] * sz;
    }
    __syncthreads();
  }
}

// ---------------------------------------------------------------------------
extern "C" void kernel_launch(void* const* d_in, const int* in_sizes, int n_in,
                              void* d_out, int out_size, void* d_ws,
                              size_t ws_size, hipStream_t stream) {
  (void)in_sizes; (void)n_in; (void)out_size; (void)ws_size;
  const float* x     = (const float*)d_in[0];
  const float* w     = (const float*)d_in[1];
  const float* gamma = (const float*)d_in[2];
  const float* beta  = (const float*)d_in[3];
  const float* xpw   = (const float*)d_in[4];
  const float* dtwp  = (const float*)d_in[5];
  const float* dtb   = (const float*)d_in[6];
  const float* alogs = (const float*)d_in[7];
  const float* Ds    = (const float*)d_in[8];
  const float* rs    = (const float*)d_in[9];
  float* out = (float*)d_out;

  float* xc    = (float*)d_ws;                       // B*L*D floats (16 MB)
  float* ps    = xc + (size_t)B_ * L_ * D_;          // NBLK1*64
  float* pq    = ps + (size_t)NBLK1 * 64;            // NBLK1*64
  float* stats = pq + (size_t)NBLK1 * 64;            // 128 floats
  float* Pw    = stats + 128;                        // NBLK3*DN
  float* Qw    = Pw + (size_t)NBLK3 * DN;            // NBLK3*DN
  float* hinit = Qw + (size_t)NBLK3 * DN;            // NBLK3*DN

  ss1d_conv_kernel<<<NBLK1, 256, 0, stream>>>(x, w, xc, ps, pq);
  ss1d_bnfin_kernel<<<1, 64, 0, stream>>>(ps, pq, gamma, beta, stats);
  ss1d_seg_kernel<<<NBLK3, 256, 0, stream>>>(xc, stats, xpw, dtwp, dtb,
                                             alogs, Pw, Qw);
  ss1d_comb_kernel<<<(B_ * DN) / 256, 256, 0, stream>>>(Pw, Qw, hinit);
  ss1d_ssm_kernel<<<NBLK3, 256, 0, stream>>>(x, xc, stats, xpw, dtwp, dtb,
                                             alogs, Ds, rs, hinit, out);
}